// emdModule_5669356836196
// MI455X (gfx1250) — compile-verified
//
#include <hip/hip_runtime.h>
#include <hip/hip_bf16.h>

typedef float v2f __attribute__((ext_vector_type(2)));
typedef float v8f __attribute__((ext_vector_type(8)));

#define NPTS   2048
#define TILE   16
#define NTILES (NPTS / TILE)   // 128
#define ITERS  20

// ---------------------------------------------------------------------------
// One-time state init: price = 0, inv = -1
// ---------------------------------------------------------------------------
__global__ void k_init_state(float* __restrict__ price, int* __restrict__ inv, int total) {
    int t = blockIdx.x * blockDim.x + threadIdx.x;
    if (t < total) { price[t] = 0.0f; inv[t] = -1; }
}

// Per-iteration clear: assign = -1, max_incr = -inf (bits), winner = NPTS
__global__ void k_clear(int* __restrict__ assign, unsigned int* __restrict__ max_bits,
                        int* __restrict__ winner, int total) {
    int t = blockIdx.x * blockDim.x + threadIdx.x;
    if (t < total) {
        assign[t]   = -1;
        max_bits[t] = 0xFF800000u;   // -inf as f32 bits; very negative as signed int
        winner[t]   = NPTS;
    }
}

// assign[b, inv[b,j]] = j  (duplicates resolved deterministically via atomicMax)
__global__ void k_scatter_assign(const int* __restrict__ inv, int* __restrict__ assign, int total) {
    int t = blockIdx.x * blockDim.x + threadIdx.x;
    if (t >= total) return;
    int b = t / NPTS;
    int j = t % NPTS;
    int i = inv[t];
    if (i >= 0) atomicMax(&assign[b * NPTS + i], j);
}

// ---------------------------------------------------------------------------
// Bidding phase: one wave per (batch, 16-row tile of points).
// value[i][j] = 2*a.b - |a|^2 - |b|^2 - price[j] via a single
// v_wmma_f32_16x16x4_f32 per 16x16 tile:
//   A row i = (x1, y1, z1, 1)
//   B col j = (2x2, 2y2, 2z2, -(|b_j|^2 + price[j]))
//   C init  = -|a_i|^2
// D layout: each lane holds column (lane&15) of rows k + (lane>=16 ? 8 : 0).
// Each lane keeps a branchless per-row top-2 over its own 128 columns; a
// final shfl_xor butterfly merges the 16 lanes of each row. No LDS, no
// barriers, no branches in the hot loop.
// ---------------------------------------------------------------------------
__global__ __launch_bounds__(32)
void k_bidding(const float* __restrict__ xyz1, const float* __restrict__ xyz2,
               const float* __restrict__ price, const int* __restrict__ assign,
               int* __restrict__ bid, float* __restrict__ score,
               const float* __restrict__ eps_p) {
    const int lane = threadIdx.x;       // 0..31 (wave32)
    const int bt   = blockIdx.x;
    const int b    = bt / NTILES;
    const int it   = bt % NTILES;
    const int i0   = it * TILE;
    const int r    = lane & 15;
    const bool hi  = lane >= 16;

    const float* x1 = xyz1  + (size_t)b * NPTS * 3;
    const float* x2 = xyz2  + (size_t)b * NPTS * 3;
    const float* pr = price + (size_t)b * NPTS;

    // --- A tile (16x4 f32): lanes 0-15 hold K=0,1 ; lanes 16-31 hold K=2,3
    const int ip = i0 + r;
    float ax = x1[ip * 3 + 0];
    float ay = x1[ip * 3 + 1];
    float az = x1[ip * 3 + 2];
    v2f A;
    A.x = hi ? az : ax;      // K=2 : K=0
    A.y = hi ? 1.0f : ay;    // K=3 : K=1

    // --- C init: c[k] = -|a_row|^2 , row = k + (hi ? 8 : 0), column = lane&15
    v8f C;
#pragma unroll
    for (int k = 0; k < 8; ++k) {
        int row = k + (hi ? 8 : 0);
        float px = x1[(i0 + row) * 3 + 0];
        float py = x1[(i0 + row) * 3 + 1];
        float pz = x1[(i0 + row) * 3 + 2];
        C[k] = -(px * px + py * py + pz * pz);
    }

    // --- per-lane, per-row top-2 state (this lane covers columns j = j0 + r)
    float b1[8], b2[8];
    int   bj[8];
#pragma unroll
    for (int k = 0; k < 8; ++k) { b1[k] = -INFINITY; b2[k] = -INFINITY; bj[k] = 0x7FFFFFFF; }

#pragma unroll 2
    for (int jt = 0; jt < NTILES; ++jt) {
        const int j0 = jt * TILE;
        const int j  = j0 + r;
        float bx = x2[j * 3 + 0];
        float by = x2[j * 3 + 1];
        float bz = x2[j * 3 + 2];
        float n2 = bx * bx + by * by + bz * bz;
        float pj = pr[j];
        v2f Bv;
        Bv.x = hi ? (2.0f * bz) : (2.0f * bx);      // B rows K=2 : K=0
        Bv.y = hi ? (-(n2 + pj)) : (2.0f * by);     // B rows K=3 : K=1

        v8f D = __builtin_amdgcn_wmma_f32_16x16x4_f32(
            /*neg_a=*/false, A, /*neg_b=*/false, Bv,
            /*c_mod=*/(short)0, C, /*reuse_a=*/false, /*reuse_b=*/false);

        // Branchless top-2 update (keeps smallest j on ties: strict >)
#pragma unroll
        for (int k = 0; k < 8; ++k) {
            float v  = D[k];
            bool  gt = v > b1[k];
            b2[k] = fmaxf(b2[k], fminf(v, b1[k]));
            bj[k] = gt ? j : bj[k];
            b1[k] = fmaxf(b1[k], v);
        }
    }

    // --- butterfly merge across the 16 lanes that share each row
#pragma unroll
    for (int m = 1; m <= 8; m <<= 1) {
#pragma unroll
        for (int k = 0; k < 8; ++k) {
            float o1 = __shfl_xor(b1[k], m, 32);
            float o2 = __shfl_xor(b2[k], m, 32);
            int   oj = __shfl_xor(bj[k], m, 32);
            bool take = (o1 > b1[k]) || ((o1 == b1[k]) && (oj < bj[k]));
            b2[k] = fmaxf(fminf(o1, b1[k]), fmaxf(o2, b2[k]));
            b1[k] = fmaxf(o1, b1[k]);
            bj[k] = take ? oj : bj[k];
        }
    }

    // --- lane (lane&15)==k publishes row k + (hi?8:0)
    const float eps = *eps_p;
#pragma unroll
    for (int k = 0; k < 8; ++k) {
        if (r == k) {
            int row = k + (hi ? 8 : 0);
            size_t gi = (size_t)b * NPTS + (i0 + row);
            float incr = b1[k] - b2[k] + eps;       // always >= eps > 0
            int   a    = assign[gi];
            bid[gi]    = bj[k];
            score[gi]  = (a < 0) ? incr : -INFINITY;
        }
    }
}

// Per-object max bid: atomicMax on f32 bit pattern (scores > 0, init is -inf)
__global__ void k_reduce_max(const int* __restrict__ bid, const float* __restrict__ score,
                             unsigned int* __restrict__ max_bits, int total) {
    int t = blockIdx.x * blockDim.x + threadIdx.x;
    if (t >= total) return;
    float s = score[t];
    if (s == -INFINITY) return;
    int b = t / NPTS;
    int j = bid[t];
    atomicMax((int*)&max_bits[b * NPTS + j], __float_as_int(s));
}

// Winner = min point index among bidders achieving the max bid
__global__ void k_select_winner(const int* __restrict__ bid, const float* __restrict__ score,
                                const unsigned int* __restrict__ max_bits,
                                int* __restrict__ winner, int total) {
    int t = blockIdx.x * blockDim.x + threadIdx.x;
    if (t >= total) return;
    float s = score[t];
    if (s == -INFINITY) return;
    int b = t / NPTS;
    int i = t % NPTS;
    int j = bid[t];
    float mx = __int_as_float((int)max_bits[b * NPTS + j]);
    if (s >= mx) atomicMin(&winner[b * NPTS + j], i);
}

// Commit: objects with a winner update inv and raise their price
__global__ void k_update(int* __restrict__ inv, float* __restrict__ price,
                         const int* __restrict__ winner,
                         const unsigned int* __restrict__ max_bits, int total) {
    int t = blockIdx.x * blockDim.x + threadIdx.x;
    if (t >= total) return;
    int w = winner[t];
    if (w < NPTS) {
        inv[t]    = w;
        price[t] += __int_as_float((int)max_bits[t]);
    }
}

// Final distances + assignment output
__global__ void k_finalize(const float* __restrict__ xyz1, const float* __restrict__ xyz2,
                           const int* __restrict__ assign,
                           float* __restrict__ dist_out, int* __restrict__ assign_out, int total) {
    int t = blockIdx.x * blockDim.x + threadIdx.x;
    if (t >= total) return;
    int b = t / NPTS;
    int i = t % NPTS;
    int a = assign[t];
    assign_out[t] = a;
    float d = 0.0f;
    if (a >= 0) {
        const float* p = xyz1 + ((size_t)b * NPTS + i) * 3;
        const float* q = xyz2 + ((size_t)b * NPTS + a) * 3;
        float dx = p[0] - q[0], dy = p[1] - q[1], dz = p[2] - q[2];
        d = dx * dx + dy * dy + dz * dz;
    }
    dist_out[t] = d;
}

// ---------------------------------------------------------------------------
extern "C" void kernel_launch(void* const* d_in, const int* in_sizes, int n_in,
                              void* d_out, int out_size, void* d_ws, size_t ws_size,
                              hipStream_t stream) {
    (void)n_in; (void)out_size; (void)ws_size;

    const float* xyz1 = (const float*)d_in[0];
    const float* xyz2 = (const float*)d_in[1];
    const float* eps  = (const float*)d_in[2];
    // d_in[3] = iters (20); launch count must be host-static for graph capture.

    const int B  = in_sizes[0] / (NPTS * 3);   // 8
    const int BM = B * NPTS;                   // 16384

    // Workspace carve-out (7 * BM * 4 bytes = ~448 KB)
    char* w = (char*)d_ws;
    float*        price    = (float*)w;        w += (size_t)BM * 4;
    int*          inv      = (int*)w;          w += (size_t)BM * 4;
    int*          assign   = (int*)w;          w += (size_t)BM * 4;
    int*          bid      = (int*)w;          w += (size_t)BM * 4;
    float*        score    = (float*)w;        w += (size_t)BM * 4;
    unsigned int* max_bits = (unsigned int*)w; w += (size_t)BM * 4;
    int*          winner   = (int*)w;          w += (size_t)BM * 4;

    const int thr = 256;
    const int blk = (BM + thr - 1) / thr;

    k_init_state<<<blk, thr, 0, stream>>>(price, inv, BM);

    for (int it = 0; it < ITERS; ++it) {
        k_clear<<<blk, thr, 0, stream>>>(assign, max_bits, winner, BM);
        k_scatter_assign<<<blk, thr, 0, stream>>>(inv, assign, BM);
        k_bidding<<<B * NTILES, 32, 0, stream>>>(xyz1, xyz2, price, assign, bid, score, eps);
        k_reduce_max<<<blk, thr, 0, stream>>>(bid, score, max_bits, BM);
        k_select_winner<<<blk, thr, 0, stream>>>(bid, score, max_bits, winner, BM);
        k_update<<<blk, thr, 0, stream>>>(inv, price, winner, max_bits, BM);
    }

    // Final assignment derivation + outputs: [dist (B*N f32) | assignment (B*N i32)]
    k_clear<<<blk, thr, 0, stream>>>(assign, max_bits, winner, BM);
    k_scatter_assign<<<blk, thr, 0, stream>>>(inv, assign, BM);

    float* dist_out   = (float*)d_out;
    int*   assign_out = (int*)d_out + BM;
    k_finalize<<<blk, thr, 0, stream>>>(xyz1, xyz2, assign, dist_out, assign_out, BM);
}